// DGDAGRNN_68272800137724
// MI455X (gfx1250) — compile-verified
//
#include <hip/hip_runtime.h>
#include <math.h>

#define NN 32768
#define EE 131072
#define LL 8
#define VHS 100
#define NVT 3
#define NROUNDS 2
#define PER (NN / LL)             // 4096 nodes per layer (multiple of 16)
#define G3 (3 * VHS)              // 300
#define NT_MSG ((VHS + 15) / 16)  // 7 column tiles for 100-wide message output

typedef __attribute__((ext_vector_type(2))) float v2f;
typedef __attribute__((ext_vector_type(8))) float v8f;

__device__ __forceinline__ float sigmoidf_(float x) {
    return 1.0f / (1.0f + __expf(-x));
}

// ---------------------------------------------------------------- utilities
__global__ void zero_f32(float* __restrict__ p, int n) {
    int i = blockIdx.x * blockDim.x + threadIdx.x;
    if (i < n) p[i] = 0.0f;
}

__global__ void zero_i32(int* __restrict__ p, int n) {
    int i = blockIdx.x * blockDim.x + threadIdx.x;
    if (i < n) p[i] = 0;
}

// ------------------------------------------------------- edge bucketing (device)
__global__ void count_edges(const int* __restrict__ ei, const int* __restrict__ nl,
                            int* __restrict__ cntF, int* __restrict__ cntB) {
    int e = blockIdx.x * blockDim.x + threadIdx.x;
    if (e >= EE) return;
    int s = ei[e], d = ei[EE + e];
    atomicAdd(&cntF[nl[d]], 1);                 // forward: edge feeds dst's layer
    atomicAdd(&cntB[(LL - 1) - nl[s]], 1);      // backward: reversed edge feeds src's bwd layer
}

__global__ void scan_offsets(const int* __restrict__ cntF, const int* __restrict__ cntB,
                             int* __restrict__ offF, int* __restrict__ offB,
                             int* __restrict__ curF, int* __restrict__ curB) {
    if (threadIdx.x != 0 || blockIdx.x != 0) return;
    int af = 0, ab = 0;
    for (int l = 0; l < LL; ++l) {
        offF[l] = af; af += cntF[l];
        offB[l] = ab; ab += cntB[l];
        curF[l] = 0;  curB[l] = 0;
    }
}

__global__ void scatter_edges(const int* __restrict__ ei, const int* __restrict__ nl,
                              int* __restrict__ bucketF, int* __restrict__ bucketB,
                              const int* __restrict__ offF, const int* __restrict__ offB,
                              int* __restrict__ curF, int* __restrict__ curB) {
    int e = blockIdx.x * blockDim.x + threadIdx.x;
    if (e >= EE) return;
    int s = ei[e], d = ei[EE + e];
    int lf = nl[d];
    int pf = atomicAdd(&curF[lf], 1);
    bucketF[offF[lf] + pf] = e;
    int lb = (LL - 1) - nl[s];
    int pb = atomicAdd(&curB[lb], 1);
    bucketB[offB[lb] + pb] = e;
}

// --------------------------------------------------- fp32 WMMA GEMM (K templated)
// C[PER x N] = A[PER x K] @ W[N x K]^T + bias[N]; one wave per 16x16 tile.
// lda == K for every call site. M == PER (multiple of 16): no row guards.
// Column guard only at the store; load columns are clamped (a WMMA output
// column depends only on the matching B column, so garbage never leaks).
template <int K>
__global__ void wmma_gemm_k(const float* __restrict__ A,
                            const float* __restrict__ W,
                            const float* __restrict__ bias,
                            float* __restrict__ C, int ldo, int N) {
    int lane = threadIdx.x;
    int half = lane >> 4;          // 0: K pair lo / rows 0..7, 1: K pair hi / rows 8..15
    int l16  = lane & 15;
    int mBase = blockIdx.x << 4;
    int nBase = blockIdx.y << 4;
    int row  = mBase + l16;                    // always < PER
    int col  = nBase + l16;
    int colL = (col < N) ? col : (N - 1);      // clamped load column

    const float* arow = A + (size_t)row  * K;
    const float* brow = W + (size_t)colL * K;

    v8f acc = {};
    if constexpr ((K & 3) == 0) {
#pragma unroll
        for (int k0 = 0; k0 < K; k0 += 4) {
            int kk = k0 + half * 2;            // even => 8B aligned (K even)
            v2f a = *(const v2f*)(arow + kk);
            v2f b = *(const v2f*)(brow + kk);
            acc = __builtin_amdgcn_wmma_f32_16x16x4_f32(
                false, a, false, b, (short)0, acc, false, false);
        }
    } else {                                   // K == NVT == 3: single step, pad k=3
        int kk = half * 2;
        v2f a, b;
        a.x = arow[kk];
        a.y = (kk + 1 < K) ? arow[kk + 1] : 0.0f;
        b.x = brow[kk];
        b.y = (kk + 1 < K) ? brow[kk + 1] : 0.0f;
        acc = __builtin_amdgcn_wmma_f32_16x16x4_f32(
            false, a, false, b, (short)0, acc, false, false);
    }

    float bv = bias[colL];
    if (col < N) {
#pragma unroll
        for (int v = 0; v < 8; ++v) {
            int r = mBase + v + half * 8;
            C[(size_t)r * ldo + col] = acc[v] + bv;
        }
    }
}

// ------------------------------------------- gated edge messages, one layer
// For edges in bucket[off..off+cnt): gather hj = h[gidx[e]] (VHS wide),
// G = hj @ Wg^T, Mv = hj @ Wm^T, msg = sigmoid(G + bg) * Mv,
// atomically accumulate into agg[sidx[e] - layerBase].
// Invalid tail lanes read a zeroed dummy row (no divergent loads).
__global__ void edge_msg(const float* __restrict__ h, const float* __restrict__ zrow,
                         float* __restrict__ agg,
                         const int* __restrict__ gidx, const int* __restrict__ sidx,
                         const int* __restrict__ bucket,
                         const int* __restrict__ cntPtr, const int* __restrict__ offPtr,
                         const float* __restrict__ Wg, const float* __restrict__ bg,
                         const float* __restrict__ Wm, int layerBase) {
    int cnt = *cntPtr;
    if (cnt == 0) return;
    int base = *offPtr;

    int lane = threadIdx.x;
    int half = lane >> 4;
    int l16  = lane & 15;
    int nTilesE = (cnt + 15) >> 4;
    int totalW = nTilesE * NT_MSG;

    for (int w = blockIdx.x; w < totalW; w += gridDim.x) {
        int te = w / NT_MSG;
        int tn = w % NT_MSG;

        int eSlot = te * 16 + l16;
        int myE   = (eSlot < cnt) ? bucket[base + eSlot] : -1;
        const float* arow = (myE >= 0) ? (h + (size_t)gidx[myE] * VHS) : zrow;

        int colW  = tn * 16 + l16;
        bool cOK  = colW < VHS;
        int colWL = cOK ? colW : (VHS - 1);    // clamped load column
        const float* gr = Wg + (size_t)colWL * VHS;
        const float* mr = Wm + (size_t)colWL * VHS;

        v8f accG = {}, accM = {};
#pragma unroll
        for (int k0 = 0; k0 < VHS; k0 += 4) {  // VHS = 100 = 25 * 4, exact
            int kk = k0 + half * 2;
            v2f a  = *(const v2f*)(arow + kk);
            v2f bG = *(const v2f*)(gr + kk);
            v2f bM = *(const v2f*)(mr + kk);
            accG = __builtin_amdgcn_wmma_f32_16x16x4_f32(
                false, a, false, bG, (short)0, accG, false, false);
            accM = __builtin_amdgcn_wmma_f32_16x16x4_f32(
                false, a, false, bM, (short)0, accM, false, false);
        }

        if (cOK) {
            float bgv = bg[colW];
#pragma unroll
            for (int v = 0; v < 8; ++v) {
                int slot = te * 16 + v + half * 8;
                if (slot < cnt) {
                    int e2   = bucket[base + slot];
                    int dRow = sidx[e2];
                    float g   = sigmoidf_(accG[v] + bgv);
                    float msg = g * accM[v];
                    atomicAdd(&agg[(size_t)(dRow - layerBase) * VHS + colW], msg);
                }
            }
        }
    }
}

// --------------------------------------------------------- GRU elementwise
// PyTorch GRUCell with hidden = agg:  h' = (1-z)*n + z*agg
__global__ void gru_update(const float* __restrict__ gi, const float* __restrict__ gh,
                           const float* __restrict__ agg, float* __restrict__ h,
                           int layerBase) {
    int i = blockIdx.x * blockDim.x + threadIdx.x;
    if (i >= PER * VHS) return;
    int rw = i / VHS, j = i % VHS;
    const float* gir = gi + (size_t)rw * G3;
    const float* ghr = gh + (size_t)rw * G3;
    float r  = sigmoidf_(gir[j] + ghr[j]);
    float z  = sigmoidf_(gir[VHS + j] + ghr[VHS + j]);
    float n  = tanhf(gir[2 * VHS + j] + r * ghr[2 * VHS + j]);
    float a  = agg[(size_t)rw * VHS + j];
    h[(size_t)(layerBase + rw) * VHS + j] = (1.0f - z) * n + z * a;
}

// ------------------------------------------------------------------ driver
extern "C" void kernel_launch(void* const* d_in, const int* in_sizes, int n_in,
                              void* d_out, int out_size, void* d_ws, size_t ws_size,
                              hipStream_t stream) {
    (void)in_sizes; (void)n_in; (void)out_size; (void)ws_size;

    const float* x     = (const float*)d_in[0];
    const int*   ei    = (const int*)d_in[1];   // [2, E]: src row then dst row
    const int*   nl    = (const int*)d_in[2];
    const float* Wg_f  = (const float*)d_in[3];
    const float* bg_f  = (const float*)d_in[4];
    const float* Wm_f  = (const float*)d_in[5];
    const float* Wg_b  = (const float*)d_in[6];
    const float* bg_b  = (const float*)d_in[7];
    const float* Wm_b  = (const float*)d_in[8];
    const float* Wih_f = (const float*)d_in[9];
    const float* Whh_f = (const float*)d_in[10];
    const float* bih_f = (const float*)d_in[11];
    const float* bhh_f = (const float*)d_in[12];
    const float* Wih_b = (const float*)d_in[13];
    const float* Whh_b = (const float*)d_in[14];
    const float* bih_b = (const float*)d_in[15];
    const float* bhh_b = (const float*)d_in[16];
    const float* Wp    = (const float*)d_in[17];
    const float* bp    = (const float*)d_in[18];

    float* h = (float*)d_out;                 // h lives directly in d_out (N x VHS)

    // workspace carve-up (every region an even float count -> 8B alignment kept)
    float* zrow = (float*)d_ws;               // 128 zero floats (dummy A row)
    float* agg  = zrow + 128;                 // PER * VHS
    float* gi   = agg + PER * VHS;            // PER * 300
    float* gh   = gi + PER * G3;              // PER * 300
    float* inp  = gh + PER * G3;              // PER * NVT (+4 pad, stays even)
    int* bucketF = (int*)(inp + PER * NVT + 4);  // E
    int* bucketB = bucketF + EE;              // E
    int* cntF = bucketB + EE;                 // 8
    int* cntB = cntF + LL;
    int* offF = cntB + LL;
    int* offB = offF + LL;
    int* curF = offB + LL;
    int* curB = curF + LL;

    // init: h = 0, zero row = 0, counters = 0
    zero_f32<<<(NN * VHS + 255) / 256, 256, 0, stream>>>(h, NN * VHS);
    zero_f32<<<1, 128, 0, stream>>>(zrow, 128);
    zero_i32<<<1, 64, 0, stream>>>(cntF, 2 * LL);

    // bucket edges by receiving layer (forward: dst layer, backward: src bwd layer)
    count_edges<<<EE / 256, 256, 0, stream>>>(ei, nl, cntF, cntB);
    scan_offsets<<<1, 1, 0, stream>>>(cntF, cntB, offF, offB, curF, curB);
    scatter_edges<<<EE / 256, 256, 0, stream>>>(ei, nl, bucketF, bucketB,
                                                offF, offB, curF, curB);

    const dim3 blk(32, 1, 1);
    const dim3 gridGru(PER / 16, (G3 + 15) / 16, 1);   // 256 x 19
    const dim3 gridProj(PER / 16, 1, 1);
    const int  edgeBlocks = 2048;
    const int  zeroAggBlocks = (PER * VHS + 255) / 256;
    const int  gruBlocks = (PER * VHS + 255) / 256;

    for (int r = 0; r < NROUNDS; ++r) {
        // ---------------- forward sweep ----------------
        for (int l = 0; l < LL; ++l) {
            int base = l * PER;
            const float* inpA;
            if (r == 0) {
                inpA = x + (size_t)base * NVT;
            } else {
                // inp = projector(h) for this layer's nodes  (K=100, N=3)
                wmma_gemm_k<VHS><<<gridProj, blk, 0, stream>>>(
                    h + (size_t)base * VHS, Wp, bp, inp, NVT, NVT);
                inpA = inp;
            }
            zero_f32<<<zeroAggBlocks, 256, 0, stream>>>(agg, PER * VHS);
            edge_msg<<<edgeBlocks, blk, 0, stream>>>(
                h, zrow, agg, ei /*gather src*/, ei + EE /*scatter dst*/,
                bucketF, cntF + l, offF + l, Wg_f, bg_f, Wm_f, base);
            wmma_gemm_k<NVT><<<gridGru, blk, 0, stream>>>(
                inpA, Wih_f, bih_f, gi, G3, G3);
            wmma_gemm_k<VHS><<<gridGru, blk, 0, stream>>>(
                agg, Whh_f, bhh_f, gh, G3, G3);
            gru_update<<<gruBlocks, 256, 0, stream>>>(gi, gh, agg, h, base);
        }
        // ---------------- backward sweep ----------------
        for (int l = 0; l < LL; ++l) {
            int base = (LL - 1 - l) * PER;     // nodes with nl_b == l
            zero_f32<<<zeroAggBlocks, 256, 0, stream>>>(agg, PER * VHS);
            edge_msg<<<edgeBlocks, blk, 0, stream>>>(
                h, zrow, agg, ei + EE /*gather dst*/, ei /*scatter src*/,
                bucketB, cntB + l, offB + l, Wg_b, bg_b, Wm_b, base);
            wmma_gemm_k<VHS><<<gridGru, blk, 0, stream>>>(
                h + (size_t)base * VHS, Wih_b, bih_b, gi, G3, G3);
            wmma_gemm_k<VHS><<<gridGru, blk, 0, stream>>>(
                agg, Whh_b, bhh_b, gh, G3, G3);
            gru_update<<<gruBlocks, 256, 0, stream>>>(gi, gh, agg, h, base);
        }
    }
}